// BodyAvgDiseaseFeatureAttn2_30185030156859
// MI455X (gfx1250) — compile-verified
//
#include <hip/hip_runtime.h>

// ---------------------------------------------------------------------------
// BodyAvgDiseaseFeatureAttn2 on MI455X (gfx1250, wave32, WMMA)
//
// Fused view:  out[b,o] = (1/15) * sum_s  ( x[b,s,:] . Weff[o,:] ) + bias[o]
//   Weff[o,k] = softmax_c(attn)[o, k/36] * Wfc[o,k]     (K = 576, O = 83)
// => GEMM  X[61440 x 576] * Weff^T[576 x 83], mean over 15-row groups.
// fp32 WMMA (V_WMMA_F32_16X16X4_F32): exact precision, still HBM-bound.
// ---------------------------------------------------------------------------

typedef float v2f __attribute__((ext_vector_type(2)));
typedef float v4f __attribute__((ext_vector_type(4)));
typedef float v8f __attribute__((ext_vector_type(8)));

#define B_VOL      4096
#define S_SL       15
#define C_CH       16
#define HW_SP      36
#define K_FEAT     576          // C*H*W
#define N_OUT      83
#define N_PAD      96           // 6 tiles of 16
#define NT_TILES   6
#define KK_TOT     144          // K_FEAT / 4
#define KK_CHUNK   36           // K-chunk staged in LDS (4 chunks)
#define N_CHUNKS   4
#define CHUNK_FLOATS (KK_CHUNK * NT_TILES * 32 * 2)   // 13824 floats = 54 KB
#define BFRAG_FLOATS (KK_TOT   * NT_TILES * 32 * 2)   // 55296 floats = 216 KB

// ---------------------------------------------------------------------------
// Prep: softmax attention, fuse into FC weights, emit WMMA B-fragment layout.
// B-matrix (4x16, KxN) striping for V_WMMA_F32_16X16X4_F32 (per ISA 7.12.2):
//   lane n in [0,16) holds column n; K = vgpr_idx + 2*(lane>=16)
// Fragment buffer:  bfrag[((kk*6 + ntile)*32 + lane)*2 + vgpr_idx]
// Grid covers o in [0, 96); rows 83..95 are zero-filled (full ws init).
// ---------------------------------------------------------------------------
__global__ __launch_bounds__(64) void prep_weff_kernel(
    const float* __restrict__ w_heart,   // [27,16]
    const float* __restrict__ w_lung,    // [28,16]
    const float* __restrict__ w_fc,      // [83,576]
    float* __restrict__ bfrag)           // [BFRAG_FLOATS]
{
    const int o   = blockIdx.x;          // padded disease index, [0,96)
    const int tid = threadIdx.x;

    float attn[C_CH];
    if (o < N_OUT) {
        const float* src;
        if (o < 27)      src = w_heart + o * C_CH;          // heart block
        else if (o < 55) src = w_lung  + (o - 27) * C_CH;   // left lung
        else             src = w_lung  + (o - 55) * C_CH;   // right lung
        float mx = -3.402823e38f;
        #pragma unroll
        for (int c = 0; c < C_CH; ++c) { attn[c] = src[c]; mx = fmaxf(mx, attn[c]); }
        float sum = 0.f;
        #pragma unroll
        for (int c = 0; c < C_CH; ++c) { attn[c] = __expf(attn[c] - mx); sum += attn[c]; }
        const float inv = 1.f / sum;
        #pragma unroll
        for (int c = 0; c < C_CH; ++c) attn[c] *= inv;
    }

    const float* wrow = w_fc + (size_t)o * K_FEAT;   // only dereferenced if o<83
    const int nt = o >> 4;
    const int nl = o & 15;
    for (int k = tid; k < K_FEAT; k += 64) {
        const float w = (o < N_OUT) ? wrow[k] * attn[k / HW_SP] : 0.f;
        const int kk = k >> 2, r = k & 3;
        const int lane = ((r >> 1) << 4) + nl;       // half-wave = K msb
        bfrag[(((kk * NT_TILES) + nt) * 32 + lane) * 2 + (r & 1)] = w;
    }
}

// ---------------------------------------------------------------------------
// Main GEMM: 1 wave = 2 CT volumes (2 x 16-row M-tiles: 15 slices + 1 dummy),
// all 6 N-tiles accumulated in registers (2*6*8 = 96 acc VGPRs).
// 8 waves/block share B fragments through LDS, staged in 4 chunks of 54 KB.
// ---------------------------------------------------------------------------
__global__ __launch_bounds__(256) void bodyavg_wmma_kernel(
    const float* __restrict__ x,       // [4096,15,576]
    const float* __restrict__ bfrag,   // [BFRAG_FLOATS] B fragments
    const float* __restrict__ bias,    // [83]
    float* __restrict__ out)           // [4096,83]
{
    __shared__ __align__(16) float lB[CHUNK_FLOATS];

    const int tid  = threadIdx.x;
    const int lane = tid & 31;
    const int wave = tid >> 5;
    const int half = lane >> 4;            // 0: K=0/1 | M=0..7, 1: K=2/3 | M=8..15
    const int nl   = lane & 15;
    const int rowc = (nl < 15) ? nl : 14;  // clamp dummy 16th A-row (excluded later)

    const int b0 = blockIdx.x * 16 + wave * 2;
    const int b1 = b0 + 1;
    const float* xr0 = x + (size_t)(b0 * S_SL + rowc) * K_FEAT + 2 * half;
    const float* xr1 = x + (size_t)(b1 * S_SL + rowc) * K_FEAT + 2 * half;

    v8f acc0[NT_TILES], acc1[NT_TILES];
    #pragma unroll
    for (int t = 0; t < NT_TILES; ++t) { acc0[t] = (v8f)(0.f); acc1[t] = (v8f)(0.f); }

    for (int ch = 0; ch < N_CHUNKS; ++ch) {
        __syncthreads();
        // cooperative 128-bit staged copy of this K-chunk's B fragments
        const v4f* gsrc = (const v4f*)(bfrag + (size_t)ch * CHUNK_FLOATS);
        v4f*       ldst = (v4f*)lB;
        for (int i = tid; i < CHUNK_FLOATS / 4; i += 256) ldst[i] = gsrc[i];
        if (ch + 1 < N_CHUNKS)  // warm next chunk (global_prefetch_b8)
            __builtin_prefetch(bfrag + (size_t)(ch + 1) * CHUNK_FLOATS + tid * 32, 0, 3);
        __syncthreads();

        for (int kks = 0; kks < KK_CHUNK; ++kks) {
            const int k0 = (ch * KK_CHUNK + kks) * 4;
            // A fragments: lane m holds row m, float2 = {K0+2h, K0+2h+1}
            const v2f a0 = *(const v2f*)(xr0 + k0);
            const v2f a1 = *(const v2f*)(xr1 + k0);
            const v2f* lbp = (const v2f*)lB + (size_t)(kks * NT_TILES) * 32 + lane;
            #pragma unroll
            for (int nt = 0; nt < NT_TILES; ++nt) {
                const v2f bf = lbp[nt * 32];
                acc0[nt] = __builtin_amdgcn_wmma_f32_16x16x4_f32(
                    false, a0, false, bf, (short)0, acc0[nt], false, false);
                acc1[nt] = __builtin_amdgcn_wmma_f32_16x16x4_f32(
                    false, a1, false, bf, (short)0, acc1[nt], false, false);
            }
        }
    }

    // Slice-mean + bias.  C/D layout: vgpr v, lower lanes M=v, upper M=8+v.
    // Sum M=0..14, drop dummy M=15 (= vgpr 7, upper half).
    #pragma unroll
    for (int which = 0; which < 2; ++which) {
        const int b = which ? b1 : b0;
        const v8f* acc = which ? acc1 : acc0;
        #pragma unroll
        for (int nt = 0; nt < NT_TILES; ++nt) {
            float p = acc[nt][0] + acc[nt][1] + acc[nt][2] + acc[nt][3]
                    + acc[nt][4] + acc[nt][5] + acc[nt][6];
            p += half ? 0.f : acc[nt][7];            // upper half excludes M=15
            const float tot = p + __shfl_xor(p, 16, 32);
            const int ng = nt * 16 + nl;
            if (half == 0 && ng < N_OUT)
                out[(size_t)b * N_OUT + ng] = tot * (1.f / 15.f) + bias[ng];
        }
    }
}

// ---------------------------------------------------------------------------
extern "C" void kernel_launch(void* const* d_in, const int* in_sizes, int n_in,
                              void* d_out, int out_size, void* d_ws, size_t ws_size,
                              hipStream_t stream) {
    const float* x       = (const float*)d_in[0];  // [4096,15,16,6,6]
    const float* w_heart = (const float*)d_in[1];  // [1,27,16,1,1]
    const float* w_lung  = (const float*)d_in[2];  // [1,28,16,1,1]
    const float* w_fc    = (const float*)d_in[3];  // [83,576]
    const float* bias    = (const float*)d_in[4];  // [83]
    float*       out     = (float*)d_out;          // [4096,83]
    float*       bfrag   = (float*)d_ws;           // 216 KB B-fragment buffer

    prep_weff_kernel<<<N_PAD, 64, 0, stream>>>(w_heart, w_lung, w_fc, bfrag);
    bodyavg_wmma_kernel<<<B_VOL / 16, 256, 0, stream>>>(x, bfrag, bias, out);
}